// BatchedEmbedding_64828236366062
// MI455X (gfx1250) — compile-verified
//
#include <hip/hip_runtime.h>
#include <hip/hip_bf16.h>

// Multi-LoRA embedding: out[m,t,:] = weight[x[t],:] + lora_B[m] @ lora_A[m,:,x[t]]
// M=4, B*T=8192 tokens, d=2048, r=16, V=32000, fp32 everywhere.
// Bandwidth-bound (~335MB traffic); delta matmul done with V_WMMA_F32_16X16X4_F32
// (exact fp32, r=16 -> 4 K-steps of 4). One wave32 = one 16x16 (token x d) tile.

typedef __attribute__((ext_vector_type(2))) float v2f;
typedef __attribute__((ext_vector_type(8))) float v8f;

#define LORA_V   32000
#define LORA_D   2048
#define LORA_R   16
#define LORA_BT  8192
#define LORA_M   4
#define LORA_SCALING 1.0f

// tiles: 4 m * 512 token-tiles * 128 d-tiles = 262144; 8 waves/block -> 32768 blocks
#define TILES_PER_M   65536   // 512 * 128
#define DTILES        128

__global__ __launch_bounds__(256) void lora_embed_wmma_f32(
    const int*   __restrict__ x,        // [BT]
    const float* __restrict__ weight,   // [V, D]
    const float* __restrict__ lora_A,   // [M, R, V]
    const float* __restrict__ lora_B,   // [M, D, R]
    float*       __restrict__ out)      // [M, BT, D]
{
    const int lane = threadIdx.x & 31;
    const int wave = threadIdx.x >> 5;

    const unsigned tile = blockIdx.x * 8u + wave;     // < 262144
    const int m        = tile >> 16;                  // tile / TILES_PER_M
    const unsigned rem = tile & (TILES_PER_M - 1);
    const int tokBase  = (int)(rem >> 7) * 16;        // rem / DTILES
    const int dBase    = (int)(rem & (DTILES - 1)) * 16;

    const int row  = lane & 15;   // A-frag: output row (token); B-frag: output col (d)
    const int half = lane >> 4;   // K sub-select within chunk

    // Token index for this lane's row (lanes 16..31 duplicate 0..15; L1-hit).
    const int myIdx = x[tokBase + row];

    const float* A_m = lora_A + (size_t)m * LORA_R * LORA_V;
    const float* B_m = lora_B + (size_t)m * LORA_D * LORA_R;

    v8f c = {};
    // r=16 accumulated as 4 chunks of K=4 through fp32 WMMA.
    // A 16x4 fp32 layout: lane(0-15)=row M, VGPR0=K0,VGPR1=K1; lanes 16-31: K2,K3.
    // B 4x16 fp32 layout (row-striped across lanes): VGPR0=K{0|2}, VGPR1=K{1|3}.
    #pragma unroll
    for (int ck = 0; ck < 4; ++ck) {
        const int k0 = ck * 4 + half * 2;
        v2f a, b;
        a.x = A_m[(size_t)(k0 + 0) * LORA_V + myIdx];   // after_A[m, k0,   x[row]]
        a.y = A_m[(size_t)(k0 + 1) * LORA_V + myIdx];   // after_A[m, k0+1, x[row]]
        // lora_B[m, dBase+row, k0 : k0+2] is contiguous (stride-1 in r)
        const float* bp = B_m + (size_t)(dBase + row) * LORA_R + k0;
        b.x = bp[0];
        b.y = bp[1];
        c = __builtin_amdgcn_wmma_f32_16x16x4_f32(
                /*neg_a=*/false, a, /*neg_b=*/false, b,
                /*c_mod=*/(short)0, c, /*reuse_a=*/false, /*reuse_b=*/false);
    }

    // C/D 16x16 fp32 layout: lane col N = lane&15; VGPR i -> row M = i + 8*half.
    const int col = row;
    float* out_m = out + (size_t)m * LORA_BT * LORA_D;
    #pragma unroll
    for (int i = 0; i < 8; ++i) {
        const int trow = i + 8 * half;
        const int tIdx = __shfl(myIdx, trow, 32);                 // broadcast token idx
        const float base = weight[(size_t)tIdx * LORA_D + dBase + col];
        const float val  = base + LORA_SCALING * c[i];
        // Output is 268MB streamed once: bypass temporal caching.
        __builtin_nontemporal_store(
            val, out_m + (size_t)(tokBase + trow) * LORA_D + dBase + col);
    }
}

extern "C" void kernel_launch(void* const* d_in, const int* in_sizes, int n_in,
                              void* d_out, int out_size, void* d_ws, size_t ws_size,
                              hipStream_t stream) {
    const int*   x      = (const int*)  d_in[0];
    const float* weight = (const float*)d_in[1];
    const float* lora_A = (const float*)d_in[2];
    const float* lora_B = (const float*)d_in[3];
    float*       out    = (float*)d_out;

    // 262144 wave-tiles / 8 waves per block
    dim3 grid(32768), block(256);
    hipLaunchKernelGGL(lora_embed_wmma_f32, grid, block, 0, stream,
                       x, weight, lora_A, lora_B, out);
}